// GNN_GCNConv_homogen_33526514713237
// MI455X (gfx1250) — compile-verified
//
#include <hip/hip_runtime.h>

typedef __attribute__((ext_vector_type(2))) float v2f;
typedef __attribute__((ext_vector_type(8))) float v8f;

// ---------------------------------------------------------------------------
// WMMA fp32 GEMM: C[M,N] = A[M,K] * W[K,N] (+bias) (+relu)
// One wave per 80x16 output strip: 5 independent 16x16 accumulators (MB=5).
// Per K-step (K+=4): 2 B loads shared across 5 WMMAs, 5 A loads.
// Independent accumulators -> no WMMA->WMMA RAW chain, XDL stays pipelined.
//
// V_WMMA_F32_16X16X4_F32 operand layout (ISA 7.12.2):
//  A 16x4: lane l<16 holds A[m=l][k0],A[l][k0+1]; lane>=16 holds A[l-16][k0+2..3]
//  B 4x16: lane l<16 holds B[k0][n=l],B[k0+1][l]; lane>=16 holds B[k0+2..3][l-16]
//  C/D:    vgpr j, lane l<16 -> C[j][l]; lane>=16 -> C[j+8][l-16]
// ---------------------------------------------------------------------------
#define MB 5

__global__ __launch_bounds__(256) void k_gemm_f32_wmma(
    const float* __restrict__ A, const float* __restrict__ W,
    const float* __restrict__ bias, float* __restrict__ C,
    int M, int K, int N, int relu)
{
  const int lane   = threadIdx.x & 31;
  const int waveIb = threadIdx.x >> 5;
  const int waveId = blockIdx.x * (blockDim.x >> 5) + waveIb;
  const int tilesN = N >> 4;
  const int mGroups = (M + 16 * MB - 1) / (16 * MB);
  if (waveId >= mGroups * tilesN) return;    // wave-uniform: EXEC stays all-1

  const int tm0 = (waveId / tilesN) * (16 * MB);
  const int tn  = (waveId % tilesN) << 4;
  const int l16 = lane & 15;
  const int hi  = lane >> 4;                 // 0: lanes 0-15, 1: lanes 16-31

  // Per-subtile A row pointers (rows clamped for a possible ragged last strip)
  const float* arow[MB];
#pragma unroll
  for (int j = 0; j < MB; ++j) {
    int r = tm0 + 16 * j + l16;
    if (r > M - 1) r = M - 1;
    arow[j] = A + (size_t)r * K + 2 * hi;    // k-offset 0 or 2 by half-wave
  }
  const float* bcol = W + (size_t)(2 * hi) * N + (tn + l16);

  v8f acc[MB];
#pragma unroll
  for (int j = 0; j < MB; ++j) acc[j] = (v8f){};

  for (int k0 = 0; k0 < K; k0 += 4) {
    v2f b;
    b.x = bcol[(size_t)k0 * N];
    b.y = bcol[(size_t)(k0 + 1) * N];
#pragma unroll
    for (int j = 0; j < MB; ++j) {
      v2f a = *(const v2f*)(arow[j] + k0);   // A[row][k0+2hi], A[row][k0+2hi+1]
      acc[j] = __builtin_amdgcn_wmma_f32_16x16x4_f32(
          /*neg_a=*/false, a, /*neg_b=*/false, b,
          /*c_mod=*/(short)0, acc[j], /*reuse_a=*/false, /*reuse_b=*/false);
    }
  }

#pragma unroll
  for (int j = 0; j < MB; ++j) {
#pragma unroll
    for (int jj = 0; jj < 8; ++jj) {
      const int row  = tm0 + 16 * j + jj + 8 * hi;
      const int colg = tn + l16;
      if (row < M) {
        float v = acc[j][jj];
        if (bias) v += bias[colg];
        if (relu) v = fmaxf(v, 0.0f);
        C[(size_t)row * N + colg] = v;
      }
    }
  }
}

// deg[i] = 1 (self loop)
__global__ void k_deg_init(float* __restrict__ deg, int n) {
  int i = blockIdx.x * blockDim.x + threadIdx.x;
  if (i < n) deg[i] = 1.0f;
}

// deg[col[e]] += 1
__global__ void k_deg_count(const int* __restrict__ col, float* __restrict__ deg, int E) {
  int e = blockIdx.x * blockDim.x + threadIdx.x;
  if (e < E) atomicAdd(deg + col[e], 1.0f);
}

// dinv[i] = rsqrt(deg[i])   (deg >= 1 always)
__global__ void k_deg_finish(float* __restrict__ dinv, int n) {
  int i = blockIdx.x * blockDim.x + threadIdx.x;
  if (i < n) dinv[i] = rsqrtf(dinv[i]);
}

// out[i][f] = p[i][f] * dinv[i]^2   (self-loop term; also zero-initializes out)
__global__ void k_self_init(const float* __restrict__ p, const float* __restrict__ dinv,
                            float* __restrict__ out, int nF) {
  int i = blockIdx.x * blockDim.x + threadIdx.x;
  if (i >= nF) return;
  float di = dinv[i >> 6];
  out[i] = p[i] * di * di;
}

// out[c][f] += p[r][f] * dinv[r] * dinv[c]  — 64 consecutive threads per edge:
// coalesced 256B row gathers + coalesced atomic bursts into L2-resident table.
__global__ __launch_bounds__(256) void k_scatter(
    const float* __restrict__ p, const float* __restrict__ dinv,
    const int* __restrict__ rows, const int* __restrict__ cols,
    float* __restrict__ out, int E)
{
  long long tid = (long long)blockIdx.x * blockDim.x + threadIdx.x;
  int e = (int)(tid >> 6);
  int f = (int)(tid & 63);
  if (e >= E) return;
  int r = rows[e], c = cols[e];
  float v = p[(size_t)r * 64 + f] * dinv[r] * dinv[c];
  atomicAdd(out + (size_t)c * 64 + f, v);
}

// buf[i][f] += bias[f]; optional relu
__global__ void k_bias_act(float* __restrict__ buf, const float* __restrict__ bias,
                           int nF, int relu) {
  int i = blockIdx.x * blockDim.x + threadIdx.x;
  if (i >= nF) return;
  float v = buf[i] + bias[i & 63];
  if (relu) v = fmaxf(v, 0.0f);
  buf[i] = v;
}

// logits[e] = dot(t[src[e]], h[dst[e]]) + bb   — one wave (32 lanes) per edge
__global__ __launch_bounds__(256) void k_edge_dot(
    const float* __restrict__ t, const float* __restrict__ h,
    const int* __restrict__ src, const int* __restrict__ dst,
    const float* __restrict__ bb, float* __restrict__ out, int E)
{
  int wave = (int)(((long long)blockIdx.x * blockDim.x + threadIdx.x) >> 5);
  int lane = threadIdx.x & 31;
  if (wave >= E) return;
  int s = src[wave], d = dst[wave];
  const float* tr = t + (size_t)s * 64;
  const float* hr = h + (size_t)d * 64;
  float acc = tr[lane] * hr[lane] + tr[lane + 32] * hr[lane + 32];
#pragma unroll
  for (int off = 16; off > 0; off >>= 1)
    acc += __shfl_down(acc, off, 32);
  if (lane == 0) out[wave] = acc + bb[0];
}

// ---------------------------------------------------------------------------
extern "C" void kernel_launch(void* const* d_in, const int* in_sizes, int n_in,
                              void* d_out, int out_size, void* d_ws, size_t ws_size,
                              hipStream_t stream) {
  const float* x    = (const float*)d_in[0];   // [N,128]
  const int*   eidx = (const int*)d_in[1];     // [2,E_pred]
  const int*   pidx = (const int*)d_in[2];     // [2,E_pos]
  const float* W0   = (const float*)d_in[3];   // [128,128]
  const float* b0   = (const float*)d_in[4];   // [128]
  const float* W1   = (const float*)d_in[5];   // [128,64]
  const float* b1   = (const float*)d_in[6];   // [64]
  const float* W2   = (const float*)d_in[7];   // [64,64]
  const float* b2   = (const float*)d_in[8];   // [64]
  const float* Wb   = (const float*)d_in[9];   // [1,64,64]
  const float* bb   = (const float*)d_in[10];  // [1]
  float* logits = (float*)d_out;

  const int N  = in_sizes[0] / 128;
  const int Ep = in_sizes[1] / 2;              // prediction edges
  const int Es = in_sizes[2] / 2;              // pos (message) edges

  const int* prow = pidx;            // pos_edge_index[0] (source)
  const int* pcol = pidx + Es;       // pos_edge_index[1] (target)
  const int* esrc = eidx;            // edge_index[0]
  const int* edst = eidx + Ep;       // edge_index[1]

  // workspace: h0[N*128] | p[N*64] | hb[N*64] | dinv[N]
  float* h0   = (float*)d_ws;
  float* p    = h0 + (size_t)N * 128;
  float* hb   = p  + (size_t)N * 64;
  float* dinv = hb + (size_t)N * 64;

  const int BT = 256;
  auto cdiv = [](long long a, long long b) { return (int)((a + b - 1) / b); };

  const int mGroups = cdiv(N, 16 * MB);        // 80-row strips

  // ---- degrees -> dinv ----
  k_deg_init  <<<cdiv(N, BT), BT, 0, stream>>>(dinv, N);
  k_deg_count <<<cdiv(Es, BT), BT, 0, stream>>>(pcol, dinv, Es);
  k_deg_finish<<<cdiv(N, BT), BT, 0, stream>>>(dinv, N);

  // ---- h0 = x @ W0 + b0 ----
  k_gemm_f32_wmma<<<cdiv((long long)mGroups * 8, 8), BT, 0, stream>>>(
      x, W0, b0, h0, N, 128, 128, 0);

  // ---- conv1: p = h0 @ W1 ; aggregate ; +b1 ; relu -> hb ----
  k_gemm_f32_wmma<<<cdiv((long long)mGroups * 4, 8), BT, 0, stream>>>(
      h0, W1, nullptr, p, N, 128, 64, 0);
  k_self_init<<<cdiv((long long)N * 64, BT), BT, 0, stream>>>(p, dinv, hb, N * 64);
  k_scatter  <<<cdiv((long long)Es * 64, BT), BT, 0, stream>>>(p, dinv, prow, pcol, hb, Es);
  k_bias_act <<<cdiv((long long)N * 64, BT), BT, 0, stream>>>(hb, b1, N * 64, 1);

  // ---- conv2: p = hb @ W2 ; aggregate ; +b2 -> h0 (reused as h2, N*64) ----
  k_gemm_f32_wmma<<<cdiv((long long)mGroups * 4, 8), BT, 0, stream>>>(
      hb, W2, nullptr, p, N, 64, 64, 0);
  k_self_init<<<cdiv((long long)N * 64, BT), BT, 0, stream>>>(p, dinv, h0, N * 64);
  k_scatter  <<<cdiv((long long)Es * 64, BT), BT, 0, stream>>>(p, dinv, prow, pcol, h0, Es);
  k_bias_act <<<cdiv((long long)N * 64, BT), BT, 0, stream>>>(h0, b2, N * 64, 0);

  // ---- bilinear: t = h2 @ Wb[0] -> hb ; logits[e] = dot(t[src], h2[dst]) + bb ----
  k_gemm_f32_wmma<<<cdiv((long long)mGroups * 4, 8), BT, 0, stream>>>(
      h0, Wb, nullptr, hb, N, 64, 64, 0);
  k_edge_dot<<<cdiv((long long)Ep * 32, BT), BT, 0, stream>>>(
      hb, h0, esrc, edst, bb, logits, Ep);
}